// Ori_NN_obj_layer_51642686767183
// MI455X (gfx1250) — compile-verified
//
#include <hip/hip_runtime.h>

#define T_STEPS 96
#define DY 90
#define DP 80
#define DIN 170
#define HID 4096

typedef float v2f __attribute__((ext_vector_type(2)));
typedef float v8f __attribute__((ext_vector_type(8)));

// ---------------------------------------------------------------------------
// Copy y0 = J[0, :DY] into u_y row 0.
// ---------------------------------------------------------------------------
__global__ void init_y0_kernel(const float* __restrict__ J, float* __restrict__ u_y) {
    int i = threadIdx.x;
    if (i < DY) u_y[i] = J[i];
}

// ---------------------------------------------------------------------------
// Fast path: out[n] = relu(bias[n] + sum_k x[k] * W[k*HID + n]), K = N = 4096.
// Block = 256 threads = 8 waves; wave -> 16 neurons; grid = 32 blocks.
// Branch-free inner loop: 4 ds_load_b64 + 8 global_load_b32 (immediate
// offsets) + 1 global_prefetch_b8 + 4 v_wmma_f32_16x16x4_f32 per 16 K-rows.
// Two accumulators break the WMMA dependency chain.
// ---------------------------------------------------------------------------
__global__ __launch_bounds__(256)
void gemv16_wmma_hid_kernel(const float* __restrict__ W,
                            const float* __restrict__ bias,
                            const float* __restrict__ x,
                            float* __restrict__ out)
{
    __shared__ float xs[HID];
    for (int k = threadIdx.x; k < HID; k += 256) xs[k] = x[k];
    __syncthreads();

    const int wave  = threadIdx.x >> 5;              // wave32: 8 waves / block
    const int lane  = threadIdx.x & 31;
    const int n0    = (blockIdx.x * 8 + wave) * 16;  // 16 neurons per wave
    const int n     = n0 + (lane & 15);
    const int khalf = (lane >> 4) << 1;              // 0 (lanes 0-15) / 2 (16-31)

    const float bval = bias[n];
    v8f acc0, acc1;
#pragma unroll
    for (int i = 0; i < 8; ++i) { acc0[i] = bval; acc1[i] = 0.0f; }

    const float* wp = W + (size_t)khalf * HID + n;   // row kb, column n
    const float* pf = wp + (size_t)64 * HID;         // prefetch 64 rows ahead

    for (int k0 = 0; k0 < HID; k0 += 16) {
        v2f a0 = *(const v2f*)&xs[k0 +  0 + khalf];
        v2f a1 = *(const v2f*)&xs[k0 +  4 + khalf];
        v2f a2 = *(const v2f*)&xs[k0 +  8 + khalf];
        v2f a3 = *(const v2f*)&xs[k0 + 12 + khalf];

        v2f b0, b1, b2, b3;
        b0.x = wp[ 0 * HID]; b0.y = wp[ 1 * HID];
        b1.x = wp[ 4 * HID]; b1.y = wp[ 5 * HID];
        b2.x = wp[ 8 * HID]; b2.y = wp[ 9 * HID];
        b3.x = wp[12 * HID]; b3.y = wp[13 * HID];
        wp += (size_t)16 * HID;

        __builtin_prefetch(pf, 0, 1);                // global_prefetch_b8
        pf += (size_t)16 * HID;

        acc0 = __builtin_amdgcn_wmma_f32_16x16x4_f32(false, a0, false, b0, (short)0, acc0, false, false);
        acc1 = __builtin_amdgcn_wmma_f32_16x16x4_f32(false, a1, false, b1, (short)0, acc1, false, false);
        acc0 = __builtin_amdgcn_wmma_f32_16x16x4_f32(false, a2, false, b2, (short)0, acc0, false, false);
        acc1 = __builtin_amdgcn_wmma_f32_16x16x4_f32(false, a3, false, b3, (short)0, acc1, false, false);
    }

    const float r = fmaxf(acc0[0] + acc1[0], 0.0f);  // mid layers are always relu
    if (lane < 16) out[n] = r;
}

// ---------------------------------------------------------------------------
// Generic path (L1: K=170, N=4096; L4: K=4096, N=90).
// Branch-free via address clamping: rows >= K read row K-1 but multiply the
// zero-padded LDS activation (0 * finite = 0); columns >= Nout read column
// Nout-1 and are simply never stored. Only the final store is predicated.
// ---------------------------------------------------------------------------
__global__ __launch_bounds__(256)
void gemv16_wmma_gen_kernel(const float* __restrict__ W, int ldW,
                            const float* __restrict__ bias,
                            const float* __restrict__ xA, int lenA,
                            const float* __restrict__ xB, int lenB,
                            float* __restrict__ out, int Nout, int do_relu)
{
    __shared__ float xs[HID + 8];

    const int K    = lenA + lenB;
    const int Kpad = (K + 3) & ~3;

    for (int k = threadIdx.x; k < Kpad; k += blockDim.x) {
        float v = 0.0f;
        if (k < lenA)   v = xA[k];
        else if (k < K) v = xB[k - lenA];
        xs[k] = v;
    }
    __syncthreads();

    const int wave = threadIdx.x >> 5;
    const int lane = threadIdx.x & 31;
    const int n0   = (blockIdx.x * 8 + wave) * 16;
    if (n0 >= ((Nout + 15) & ~15)) return;           // wave-uniform exit

    const int nl     = lane & 15;
    const int n      = n0 + nl;
    const int n_eff  = (n < Nout) ? n : (Nout - 1);  // clamped column
    const int khalf  = (lane >> 4) << 1;

    const float bval = (n < Nout) ? bias[n] : 0.0f;
    v8f acc;
#pragma unroll
    for (int i = 0; i < 8; ++i) acc[i] = bval;

    const float* wp   = W + (size_t)khalf * (size_t)ldW + n_eff;
    const size_t step = (size_t)4 * (size_t)ldW;

    int k0 = 0;
    for (; k0 + 4 <= K; k0 += 4) {                   // full chunks, no checks
        v2f a = *(const v2f*)&xs[k0 + khalf];
        v2f b;
        b.x = wp[0];
        b.y = wp[ldW];
        wp += step;
        acc = __builtin_amdgcn_wmma_f32_16x16x4_f32(false, a, false, b, (short)0, acc, false, false);
    }
    if (k0 < K) {                                    // single clamped tail chunk
        v2f a = *(const v2f*)&xs[k0 + khalf];
        int r0 = k0 + khalf;     if (r0 > K - 1) r0 = K - 1;
        int r1 = k0 + khalf + 1; if (r1 > K - 1) r1 = K - 1;
        v2f b;
        b.x = W[(size_t)r0 * (size_t)ldW + n_eff];
        b.y = W[(size_t)r1 * (size_t)ldW + n_eff];
        acc = __builtin_amdgcn_wmma_f32_16x16x4_f32(false, a, false, b, (short)0, acc, false, false);
    }

    float r = acc[0];
    if (do_relu) r = fmaxf(r, 0.0f);
    if (lane < 16 && n < Nout) out[n] = r;
}

// ---------------------------------------------------------------------------
// Elementwise objectives (WEI_Y = WEI_P = 1.0).
// out layout: [ObjYupp (T*DY), ObjYlow (T*DY), ObjP1 (T*DP), ObjP2upp, ObjP2low]
// ---------------------------------------------------------------------------
__global__ __launch_bounds__(256)
void objective_kernel(const float* __restrict__ u_y, const float* __restrict__ J,
                      const float* __restrict__ Pp,
                      const float* __restrict__ UppY, const float* __restrict__ LowY,
                      const float* __restrict__ UppP, const float* __restrict__ LowP,
                      const float* __restrict__ scaleY, const float* __restrict__ minY,
                      const float* __restrict__ scaleP, const float* __restrict__ minP,
                      float* __restrict__ out)
{
    const int i  = blockIdx.x * blockDim.x + threadIdx.x;
    const int NY = T_STEPS * DY;   // 8640
    const int NP = T_STEPS * DP;   // 7680

    if (i < NY) {
        int d = i % DY;
        float yh = u_y[i] * scaleY[d] + minY[d];
        out[i]      = fmaxf(yh - UppY[i], 0.0f);
        out[NY + i] = fmaxf(LowY[i] - yh, 0.0f);
    }
    if (i < NP) {
        int t = i / DP, d = i % DP;
        float ph = J[(size_t)t * DIN + DY + d] * scaleP[d] + minP[d];
        out[2 * NY + i]          = ph * Pp[i];
        out[2 * NY + NP + i]     = fmaxf(ph - UppP[i], 0.0f);
        out[2 * NY + 2 * NP + i] = fmaxf(LowP[i] - ph, 0.0f);
    }
}

// ---------------------------------------------------------------------------
extern "C" void kernel_launch(void* const* d_in, const int* in_sizes, int n_in,
                              void* d_out, int out_size, void* d_ws, size_t ws_size,
                              hipStream_t stream)
{
    const float* J      = (const float*)d_in[0];
    const float* Pp     = (const float*)d_in[1];
    const float* UppY   = (const float*)d_in[2];
    const float* LowY   = (const float*)d_in[3];
    const float* UppP   = (const float*)d_in[4];
    const float* LowP   = (const float*)d_in[5];
    const float* W1     = (const float*)d_in[6];
    const float* b1     = (const float*)d_in[7];
    const float* W2     = (const float*)d_in[8];
    const float* b2     = (const float*)d_in[9];
    const float* W3     = (const float*)d_in[10];
    const float* b3     = (const float*)d_in[11];
    const float* W4     = (const float*)d_in[12];
    const float* b4     = (const float*)d_in[13];
    const float* scaleY = (const float*)d_in[14];
    const float* minY   = (const float*)d_in[15];
    const float* scaleP = (const float*)d_in[16];
    const float* minP   = (const float*)d_in[17];
    float* out = (float*)d_out;

    // Workspace: u_y (T*DY) | h1 (HID) | h2 (HID)
    float* u_y = (float*)d_ws;
    float* h1  = u_y + T_STEPS * DY;
    float* h2  = h1 + HID;

    init_y0_kernel<<<1, 128, 0, stream>>>(J, u_y);

    const dim3 blk(256);
    for (int t = 0; t < T_STEPS - 1; ++t) {
        const float* y  = u_y + (size_t)t * DY;
        const float* ut = J + (size_t)t * DIN + DY;

        // L1: h1 = relu([y | ut] @ W1 + b1)      K=170, N=4096
        gemv16_wmma_gen_kernel<<<32, blk, 0, stream>>>(W1, HID, b1, y, DY, ut, DIN - DY, h1, HID, 1);
        // L2: h2 = relu(h1 @ W2 + b2)            K=4096, N=4096 (fast path)
        gemv16_wmma_hid_kernel<<<32, blk, 0, stream>>>(W2, b2, h1, h2);
        // L3: h1 = relu(h2 @ W3 + b3)            K=4096, N=4096 (fast path)
        gemv16_wmma_hid_kernel<<<32, blk, 0, stream>>>(W3, b3, h2, h1);
        // L4: y_{t+1} = h1 @ W4 + b4             K=4096, N=90
        gemv16_wmma_gen_kernel<<<1, blk, 0, stream>>>(W4, DY, b4, h1, HID, nullptr, 0,
                                                      u_y + (size_t)(t + 1) * DY, DY, 0);
    }

    const int NY = T_STEPS * DY;                 // 8640 >= T*DP
    objective_kernel<<<(NY + 255) / 256, 256, 0, stream>>>(
        u_y, J, Pp, UppY, LowY, UppP, LowP, scaleY, minY, scaleP, minP, out);
}